// GroupBy_40939628265915
// MI455X (gfx1250) — compile-verified
//
#include <hip/hip_runtime.h>

// Reference shapes: unary (N,128) f32, binary (E,64) f32 (unused),
// deltas (E,320) f32, index1 (E,1) i32, index2 (E,1) i32.
// out = zeros(N,128).at[i1].add(deltas[:, :128]).at[i2].add(deltas[:,128:256])
// b   = deltas[:, 256:320]
// d_out = [out (N*128 floats)][b (E*64 floats)]

#define NU 128          // N_UNARY
#define NB 64           // binary feature width
#define DROW (2*NU+NB)  // 320 floats per deltas row

__global__ __launch_bounds__(256) void zero_f4(float4* __restrict__ p, long n4) {
  long i = (long)blockIdx.x * blockDim.x + threadIdx.x;
  const long stride = (long)gridDim.x * blockDim.x;
  const float4 z = make_float4(0.f, 0.f, 0.f, 0.f);
  for (; i < n4; i += stride) p[i] = z;
}

__global__ __launch_bounds__(256) void scatter_add_kernel(
    const float* __restrict__ deltas,
    const int*   __restrict__ index1,
    const int*   __restrict__ index2,
    float*       __restrict__ out,    // N x 128, zero-initialized
    float*       __restrict__ bout,   // E x 64
    int E) {
  const int lane = threadIdx.x & 31;        // wave32
  const int wid  = threadIdx.x >> 5;        // wave id in block (0..7)
  const int wpb  = blockDim.x >> 5;         // waves per block
  const long wstride = (long)gridDim.x * wpb;

  for (long e = (long)blockIdx.x * wpb + wid; e < E; e += wstride) {
    const float* drow = deltas + e * (long)DROW;

    // Prefetch next iteration's 1280-byte deltas row (10 x 128B cachelines).
    // Lowers to global_prefetch_b8 on gfx1250; hides HBM latency.
    long en = e + wstride;
    if (en < E && lane < 10)
      __builtin_prefetch(deltas + en * (long)DROW + (long)lane * 32, 0, 1);

    const int i1 = index1[e];
    const int i2 = index2[e];
    float* r1 = out + (long)i1 * NU;
    float* r2 = out + (long)i2 * NU;

    // Each atomic/load instruction covers 128 contiguous bytes across the
    // wave (lane l -> column j*32+l) so L2 atomic units see whole-line RMWs.
#pragma unroll
    for (int j = 0; j < 4; ++j) {
      const int c = j * 32 + lane;
      // NT loads: deltas is streamed once; keep `out` resident in 192MB L2.
      const float ux = __builtin_nontemporal_load(drow + c);
      const float uy = __builtin_nontemporal_load(drow + NU + c);
      // Hardware global_atomic_add_f32, no-return (STOREcnt path).
      unsafeAtomicAdd(r1 + c, ux);
      unsafeAtomicAdd(r2 + c, uy);
    }

    // b passthrough: streaming copy, NT both ways to avoid L2 pollution.
#pragma unroll
    for (int j = 0; j < 2; ++j) {
      const int c = j * 32 + lane;
      const float bv = __builtin_nontemporal_load(drow + 2 * NU + c);
      __builtin_nontemporal_store(bv, bout + e * (long)NB + c);
    }
  }
}

extern "C" void kernel_launch(void* const* d_in, const int* in_sizes, int n_in,
                              void* d_out, int out_size, void* d_ws, size_t ws_size,
                              hipStream_t stream) {
  (void)n_in; (void)d_ws; (void)ws_size; (void)out_size;
  // inputs: 0=unary (N*128), 1=binary (E*64, unused), 2=deltas (E*320),
  //         3=index1 (E), 4=index2 (E)
  const float* deltas = (const float*)d_in[2];
  const int*   index1 = (const int*)d_in[3];
  const int*   index2 = (const int*)d_in[4];

  const long N = (long)in_sizes[0] / NU;
  const long E = (long)in_sizes[3];

  float* out  = (float*)d_out;
  float* bout = out + N * NU;

  // 1) Zero the accumulation region (d_out is poisoned before timing).
  const long n4 = (N * NU) / 4;
  zero_f4<<<4096, 256, 0, stream>>>((float4*)out, n4);

  // 2) Fused scatter-add + b copy. 4096 blocks x 8 waves = 32768 waves,
  //    ~15 edge rows per wave via grid-stride loop.
  scatter_add_kernel<<<4096, 256, 0, stream>>>(deltas, index1, index2,
                                               out, bout, (int)E);
}